// PN2SSG_52802327937623
// MI455X (gfx1250) — compile-verified
//
#include <hip/hip_runtime.h>
#include <hip/hip_bf16.h>
#include <stdint.h>

#define EPS_BN 1e-5f

typedef float v2f __attribute__((ext_vector_type(2)));
typedef float v8f __attribute__((ext_vector_type(8)));

struct LayerP {
    const float* w;    // (cout, cin) row-major (possibly padded/permuted)
    const float* sc;   // folded BN scale  = g * rsqrt(rv + eps)
    const float* bi;   // folded BN bias   = b - rm * sc
};

struct BNSrc { const float* g; const float* b; const float* rm; const float* rv; int cout; int off; };
struct BNPack { BNSrc l[9]; };

// ---------------------------------------------------------------------------
// WMMA f32 16x16x4 : D = A(16x4) * B(4x16) + C(16x16), exact f32 math.
// A layout (ISA 7.12.2): lanes 0-15 hold rows M=0..15 with VGPR0=K0,VGPR1=K1;
// lanes 16-31 same rows with K2/K3.  B mirrors the C/D "row striped across
// lanes" pattern: VGPR v, lanes 0-15 -> K=v,N=lane; lanes 16-31 -> K=v+2.
// ---------------------------------------------------------------------------
__device__ __forceinline__ v8f wmma4(v2f a, v2f b, v8f c) {
    return __builtin_amdgcn_wmma_f32_16x16x4_f32(
        /*neg_a=*/false, a, /*neg_b=*/false, b,
        /*c_mod=*/(short)0, c, /*reuse_a=*/false, /*reuse_b=*/false);
}

// One 16x16 output tile: rows m0..m0+15 of X (leading dim ldx, K padded to a
// multiple of 4), cols n0..n0+15 of W (cout x ldw row-major, zero-padded).
// All K-strides are multiples of 4 floats -> v2f (b64) loads are aligned.
__device__ __forceinline__ v8f gemm_tile(const float* __restrict__ X, int ldx, int m0,
                                         const float* __restrict__ W, int ldw,
                                         int n0, int kdim, int lane) {
    v8f acc = {0.f, 0.f, 0.f, 0.f, 0.f, 0.f, 0.f, 0.f};
    const int l = lane & 15;
    const int half = lane >> 4;
    const float* xr = X + (size_t)(m0 + l) * ldx + 2 * half;
    const float* wr = W + (size_t)(n0 + l) * ldw + 2 * half;
    __builtin_prefetch(wr, 0, 3);   // global_prefetch_b8, keep near
    for (int k0 = 0; k0 < kdim; k0 += 4) {
        v2f a  = *(const v2f*)(xr + k0);
        v2f bb = *(const v2f*)(wr + k0);
        acc = wmma4(a, bb, acc);
    }
    return acc;
}

// BN+ReLU epilogue -> LDS.  C/D layout: VGPR v, lanes 0-15 = (M=v, N=lane),
// lanes 16-31 = (M=v+8, N=lane-16).
__device__ __forceinline__ void epilogue_to_lds(v8f acc, float* Y, int ldy, int m0, int n0,
                                                LayerP L, int lane) {
    const int l = lane & 15, half = lane >> 4;
    const int ch = n0 + l;
    const float sc = L.sc[ch];
    const float bi = L.bi[ch];
#pragma unroll
    for (int v = 0; v < 8; ++v) {
        float y = fmaxf(acc[v] * sc + bi, 0.f);
        Y[(size_t)(m0 + v + 8 * half) * ldy + ch] = y;
    }
}

// BN+ReLU + max-pool-over-rows epilogue -> global, via int atomicMax on the
// float bit pattern (valid: ReLU output >= 0, buffer zero-initialized).
__device__ __forceinline__ void epilogue_maxpool(v8f acc, float* outp, int n0,
                                                 LayerP L, int lane) {
    const int l = lane & 15;
    const int ch = n0 + l;
    const float sc = L.sc[ch];
    const float bi = L.bi[ch];
    float m = 0.f;
#pragma unroll
    for (int v = 0; v < 8; ++v) m = fmaxf(m, fmaxf(acc[v] * sc + bi, 0.f));
    atomicMax((int*)(outp + ch), __float_as_int(m));
}

// ---------------------------------------------------------------------------
// Prep: (a) fold BN params into per-channel scale/bias for all 9 layers
// (2560 channels total); (b) zero-pad SA1 layer1 weights (64x3 -> 64x4) and
// pad+permute SA2 layer1 (128x131 -> 128x132, columns [feat(128)|xyz(3)|0])
// so every GEMM K-loop is branch-free and b64/b128 aligned.
// ---------------------------------------------------------------------------
__global__ void prep_kernel(BNPack P, float* __restrict__ sc, float* __restrict__ bi,
                            const float* __restrict__ w_sa1,  // 64x3
                            const float* __restrict__ w_sa2,  // 128x131
                            float* __restrict__ p1,           // 64x4
                            float* __restrict__ p2) {         // 128x132
    int i = blockIdx.x * blockDim.x + threadIdx.x;
    if (i < 2560) {
#pragma unroll
        for (int L = 0; L < 9; ++L) {
            int o0 = P.l[L].off, o1 = o0 + P.l[L].cout;
            if (i >= o0 && i < o1) {
                int ch = i - o0;
                float s = P.l[L].g[ch] * rsqrtf(P.l[L].rv[ch] + EPS_BN);
                sc[i] = s;
                bi[i] = P.l[L].b[ch] - P.l[L].rm[ch] * s;
            }
        }
    }
    int j = i - 2560;
    if (j >= 0 && j < 256) {
        int o = j >> 2, k = j & 3;
        p1[j] = (k < 3) ? w_sa1[o * 3 + k] : 0.f;
    }
    int q = i - 2560 - 256;
    if (q >= 0 && q < 128 * 132) {
        int o = q / 132, k = q - o * 132;
        float v = 0.f;
        if (k < 128)      v = w_sa2[o * 131 + 3 + k];     // feature columns
        else if (k < 131) v = w_sa2[o * 131 + (k - 128)]; // xyz columns
        p2[q] = v;
    }
}

// ---------------------------------------------------------------------------
// points (B,3,N) -> xyz (B,N,3)
// ---------------------------------------------------------------------------
__global__ void transpose_kernel(const float* __restrict__ pts, float* __restrict__ xyz) {
    int i = blockIdx.x * blockDim.x + threadIdx.x;   // b*8192 + n
    if (i >= 16 * 8192) return;
    int b = i >> 13, n = i & 8191;
#pragma unroll
    for (int d = 0; d < 3; ++d)
        xyz[(size_t)i * 3 + d] = pts[(size_t)b * 3 * 8192 + (size_t)d * 8192 + n];
}

// ---------------------------------------------------------------------------
// Farthest point sampling: one 1024-thread block per batch, dists in LDS.
// ---------------------------------------------------------------------------
__global__ __launch_bounds__(1024) void fps_kernel(const float* __restrict__ xyz,
                                                   float* __restrict__ centers) {
    __shared__ float dists[8192];
    __shared__ float swv[32];
    __shared__ int swi[32];
    __shared__ int sidx[512];
    const int b = blockIdx.x;
    const float* P = xyz + (size_t)b * 8192 * 3;
    const int tid = threadIdx.x, lane = tid & 31, wid = tid >> 5;

    for (int i = tid; i < 8192; i += 1024) dists[i] = 3.4e38f;
    __syncthreads();

    int cur = 0;
    for (int s = 0; s < 512; ++s) {
        if (tid == 0) sidx[s] = cur;
        const float cx = P[cur * 3 + 0], cy = P[cur * 3 + 1], cz = P[cur * 3 + 2];
        float bv = -1.f;
        int bi = 0;
        for (int i = tid; i < 8192; i += 1024) {
            float dx = P[i * 3 + 0] - cx, dy = P[i * 3 + 1] - cy, dz = P[i * 3 + 2] - cz;
            float d = dx * dx + dy * dy + dz * dz;
            float dd = fminf(dists[i], d);
            dists[i] = dd;
            if (dd > bv) { bv = dd; bi = i; }
        }
        for (int off = 16; off > 0; off >>= 1) {
            float ov = __shfl_down(bv, off);
            int oi = __shfl_down(bi, off);
            if (ov > bv || (ov == bv && oi < bi)) { bv = ov; bi = oi; }
        }
        if (lane == 0) { swv[wid] = bv; swi[wid] = bi; }
        __syncthreads();
        if (wid == 0) {
            bv = swv[lane];
            bi = swi[lane];
            for (int off = 16; off > 0; off >>= 1) {
                float ov = __shfl_down(bv, off);
                int oi = __shfl_down(bi, off);
                if (ov > bv || (ov == bv && oi < bi)) { bv = ov; bi = oi; }
            }
            if (lane == 0) swi[0] = bi;
        }
        __syncthreads();
        cur = swi[0];
        __syncthreads();
    }
    for (int s = tid; s < 512; s += 1024) {
        int id = sidx[s];
        centers[((size_t)b * 512 + s) * 3 + 0] = P[id * 3 + 0];
        centers[((size_t)b * 512 + s) * 3 + 1] = P[id * 3 + 1];
        centers[((size_t)b * 512 + s) * 3 + 2] = P[id * 3 + 2];
    }
}

// ---------------------------------------------------------------------------
// Ball query: one wave per center; "first k in-radius indices, ascending,
// padded with first hit (0 if none)" — the top_k(-score) semantics.
// ---------------------------------------------------------------------------
__global__ void ballquery_kernel(const float* __restrict__ src, int N,
                                 const float* __restrict__ centers, int S, int cStrideRows,
                                 float r2, int k, int* __restrict__ outIdx, int B) {
    const int lane = threadIdx.x & 31;
    const int w = blockIdx.x * (blockDim.x >> 5) + (threadIdx.x >> 5);
    if (w >= B * S) return;
    const int b = w / S, s = w % S;
    const float* C = centers + ((size_t)b * cStrideRows + s) * 3;
    const float cx = C[0], cy = C[1], cz = C[2];
    const float* P = src + (size_t)b * N * 3;
    int* out = outIdx + (size_t)w * k;

    int count = 0, first = -1;
    for (int base = 0; base < N && count < k; base += 32) {
        int i = base + lane;
        float dx = P[i * 3 + 0] - cx, dy = P[i * 3 + 1] - cy, dz = P[i * 3 + 2] - cz;
        bool in = (dx * dx + dy * dy + dz * dz) < r2;
        unsigned m = (unsigned)__ballot(in);
        if (m != 0u && first < 0) first = base + __ffs((int)m) - 1;
        int pre = __popc(m & ((1u << lane) - 1u));
        if (in && (count + pre) < k) out[count + pre] = i;
        count += __popc(m);
    }
    if (count > k) count = k;
    if (count == 0) first = 0;
    for (int j = count + lane; j < k; j += 32) out[j] = first;
}

// ---------------------------------------------------------------------------
// SA block 1: one workgroup (4 waves) per center; 32 neighbors, 3->64->64->128
// (layer1 weights pre-padded to K=4), max-pool fused into layer-3 epilogue.
// wid goes through readfirstlane so tile loops are scalar branches and EXEC
// stays all-ones through every WMMA.
// ---------------------------------------------------------------------------
__global__ __launch_bounds__(128) void sa1_kernel(const float* __restrict__ xyz,
                                                  const float* __restrict__ centers1,
                                                  const int* __restrict__ idx1,
                                                  float* __restrict__ feat1,
                                                  LayerP L1, LayerP L2, LayerP L3) {
    __shared__ __align__(16) float X[32 * 4];
    __shared__ __align__(16) float Y1[32 * 64];
    __shared__ __align__(16) float Y2[32 * 64];
    const int c = blockIdx.x;   // b*512 + s
    const int b = c >> 9;
    const int tid = threadIdx.x, lane = tid & 31;
    const int wid = __builtin_amdgcn_readfirstlane(tid >> 5);

    {   // gather relative coords (pad K 3->4 with zero)
        int row = tid >> 2, col = tid & 3;
        int j = idx1[(size_t)c * 32 + row];
        float v = 0.f;
        if (col < 3) v = xyz[((size_t)b * 8192 + j) * 3 + col] - centers1[(size_t)c * 3 + col];
        X[row * 4 + col] = v;
    }
    __syncthreads();
    for (int t = wid; t < 8; t += 4) {
        int mi = t & 1, ni = t >> 1;
        v8f acc = gemm_tile(X, 4, mi * 16, L1.w, 4, ni * 16, 4, lane);
        epilogue_to_lds(acc, Y1, 64, mi * 16, ni * 16, L1, lane);
    }
    __syncthreads();
    for (int t = wid; t < 8; t += 4) {
        int mi = t & 1, ni = t >> 1;
        v8f acc = gemm_tile(Y1, 64, mi * 16, L2.w, 64, ni * 16, 64, lane);
        epilogue_to_lds(acc, Y2, 64, mi * 16, ni * 16, L2, lane);
    }
    __syncthreads();
    float* outp = feat1 + (size_t)c * 128;
    for (int t = wid; t < 16; t += 4) {
        int mi = t & 1, ni = t >> 1;
        v8f acc = gemm_tile(Y2, 64, mi * 16, L3.w, 64, ni * 16, 64, lane);
        epilogue_maxpool(acc, outp, ni * 16, L3, lane);
    }
}

// ---------------------------------------------------------------------------
// SA block 2: 64 neighbors as 2 chunks of 32 rows; X layout (permuted to match
// the padded layer-1 weights): [feat1(128) | rel-xyz(3) | 0], row stride 132.
// The 128-float feature gather uses per-lane CDNA5 async DMA
// (global_load_async_to_lds_b128, 16B/lane, both sides 16B aligned: row
// stride 528B = 33*16), drained with s_wait_asynccnt before the barrier.
// ---------------------------------------------------------------------------
__global__ __launch_bounds__(128) void sa2_kernel(const float* __restrict__ centers1,
                                                  const int* __restrict__ idx2,
                                                  const float* __restrict__ feat1,
                                                  float* __restrict__ feat2,
                                                  LayerP L1, LayerP L2, LayerP L3) {
    __shared__ __align__(16) float X[32 * 132];
    __shared__ __align__(16) float Y1[32 * 128];
    __shared__ __align__(16) float Y2[32 * 128];
    const int blk = blockIdx.x;
    const int h = blk & 1;
    const int c = blk >> 1;   // b*128 + s
    const int b = c >> 7, s = c & 127;
    const int tid = threadIdx.x, lane = tid & 31;
    const int wid = __builtin_amdgcn_readfirstlane(tid >> 5);
    const float* ctr = centers1 + ((size_t)b * 512 + s) * 3;

    // async DMA the 128-float feature rows straight into LDS (8 rows/wave)
    for (int r = wid; r < 32; r += 4) {
        int j = idx2[(size_t)c * 64 + h * 32 + r];
        const float* gsrc = feat1 + ((size_t)b * 512 + j) * 128 + lane * 4;
        unsigned ldsb = (unsigned)(uintptr_t)(&X[r * 132 + lane * 4]);
        asm volatile("global_load_async_to_lds_b128 %0, %1, off"
                     :: "v"(ldsb), "v"(gsrc) : "memory");
    }
    {   // xyz columns (128..130) + zero pad (131), plain stores
        int r = tid >> 2, col = tid & 3;
        int j = idx2[(size_t)c * 64 + h * 32 + r];
        float v = 0.f;
        if (col < 3) v = centers1[((size_t)b * 512 + j) * 3 + col] - ctr[col];
        X[r * 132 + 128 + col] = v;
    }
    asm volatile("s_wait_asynccnt 0" ::: "memory");
    __syncthreads();

    for (int t = wid; t < 16; t += 4) {
        int mi = t & 1, ni = t >> 1;
        v8f acc = gemm_tile(X, 132, mi * 16, L1.w, 132, ni * 16, 132, lane);
        epilogue_to_lds(acc, Y1, 128, mi * 16, ni * 16, L1, lane);
    }
    __syncthreads();
    for (int t = wid; t < 16; t += 4) {
        int mi = t & 1, ni = t >> 1;
        v8f acc = gemm_tile(Y1, 128, mi * 16, L2.w, 128, ni * 16, 128, lane);
        epilogue_to_lds(acc, Y2, 128, mi * 16, ni * 16, L2, lane);
    }
    __syncthreads();
    float* outp = feat2 + (size_t)c * 256;
    for (int t = wid; t < 32; t += 4) {
        int mi = t & 1, ni = t >> 1;
        v8f acc = gemm_tile(Y2, 128, mi * 16, L3.w, 128, ni * 16, 128, lane);
        epilogue_maxpool(acc, outp, ni * 16, L3, lane);
    }
}

// ---------------------------------------------------------------------------
// Final MLP 256->256->512->1024 over (B,128) rows, 16-row chunks, 8 waves.
// Max+argmax fused via u64 packed (valBits<<32)|(127-row) global atomicMax:
// equal values resolve to the smallest row, matching jnp.argmax.
// ---------------------------------------------------------------------------
__global__ __launch_bounds__(256) void local_kernel(const float* __restrict__ feat2,
                                                    unsigned long long* __restrict__ pmax,
                                                    LayerP L1, LayerP L2, LayerP L3) {
    __shared__ __align__(16) float Y1[16 * 256];
    __shared__ __align__(16) float Y2[16 * 512];
    const int b = blockIdx.x >> 3;
    const int chunk = blockIdx.x & 7;
    const int tid = threadIdx.x, lane = tid & 31;
    const int wid = __builtin_amdgcn_readfirstlane(tid >> 5);
    const int row0 = chunk * 16;
    const float* Xg = feat2 + ((size_t)b * 128 + row0) * 256;

    for (int t = wid; t < 16; t += 8) {
        v8f acc = gemm_tile(Xg, 256, 0, L1.w, 256, t * 16, 256, lane);
        epilogue_to_lds(acc, Y1, 256, 0, t * 16, L1, lane);
    }
    __syncthreads();
    for (int t = wid; t < 32; t += 8) {
        v8f acc = gemm_tile(Y1, 256, 0, L2.w, 256, t * 16, 256, lane);
        epilogue_to_lds(acc, Y2, 512, 0, t * 16, L2, lane);
    }
    __syncthreads();
    unsigned long long* pm = pmax + (size_t)b * 1024;
    for (int t = wid; t < 64; t += 8) {
        v8f acc = gemm_tile(Y2, 512, 0, L3.w, 512, t * 16, 512, lane);
        const int l = lane & 15, half = lane >> 4;
        const int ch = t * 16 + l;
        const float sc = L3.sc[ch];
        const float bi = L3.bi[ch];
        unsigned long long best = 0ull;
#pragma unroll
        for (int v = 0; v < 8; ++v) {
            float y = fmaxf(acc[v] * sc + bi, 0.f);
            int row = row0 + v + 8 * half;
            unsigned long long pk = ((unsigned long long)__float_as_uint(y) << 32)
                                  | (unsigned long long)(127 - row);
            if (pk > best) best = pk;
        }
        atomicMax(pm + ch, best);
    }
}

__global__ void finalize_kernel(const unsigned long long* __restrict__ pmax,
                                float* __restrict__ outFeats, int* __restrict__ outIdx) {
    int i = blockIdx.x * blockDim.x + threadIdx.x;
    if (i >= 16 * 1024) return;
    unsigned long long p = pmax[i];
    outFeats[i] = __uint_as_float((unsigned)(p >> 32));
    outIdx[i] = 127 - (int)(p & 0xFFFFFFFFull);
}

// ---------------------------------------------------------------------------
// Host driver.  Input order (setup_inputs dict insertion order):
//   d_in[0] = points (16,3,8192)
//   then sa block0 L0..L2, sa block1 L0..L2, local L0..L2,
//   each layer as (w, g, b, rm, rv) per _layer() insertion order.
// ---------------------------------------------------------------------------
extern "C" void kernel_launch(void* const* d_in, const int* in_sizes, int n_in,
                              void* d_out, int out_size, void* d_ws, size_t ws_size,
                              hipStream_t stream) {
    (void)in_sizes; (void)n_in; (void)out_size; (void)ws_size;
    const float* points = (const float*)d_in[0];

    // raw per-layer input views (w,g,b,rm,rv)
    const float* W[9]; const float* G[9]; const float* Bb[9]; const float* RM[9]; const float* RV[9];
    for (int L = 0; L < 9; ++L) {
        int base = 1 + 5 * L;
        W[L]  = (const float*)d_in[base + 0];
        G[L]  = (const float*)d_in[base + 1];
        Bb[L] = (const float*)d_in[base + 2];
        RM[L] = (const float*)d_in[base + 3];
        RV[L] = (const float*)d_in[base + 4];
    }
    const int couts[9] = {64, 64, 128, 128, 128, 256, 256, 512, 1024};
    int offs[9]; int tot = 0;
    for (int L = 0; L < 9; ++L) { offs[L] = tot; tot += couts[L]; }   // tot = 2560

    char* ws = (char*)d_ws;
    size_t off = 0;
    auto alloc = [&](size_t bytes) {
        void* p = ws + off;
        off = (off + bytes + 255) & ~(size_t)255;
        return p;
    };
    float* xyz     = (float*)alloc((size_t)16 * 8192 * 3 * sizeof(float));
    float* centers = (float*)alloc((size_t)16 * 512 * 3 * sizeof(float));
    int*   idx1    = (int*)alloc((size_t)16 * 512 * 32 * sizeof(int));
    float* feat1   = (float*)alloc((size_t)16 * 512 * 128 * sizeof(float));
    int*   idx2    = (int*)alloc((size_t)16 * 128 * 64 * sizeof(int));
    float* feat2   = (float*)alloc((size_t)16 * 128 * 256 * sizeof(float));
    unsigned long long* pmax = (unsigned long long*)alloc((size_t)16 * 1024 * 8);
    float* w1p_sa1 = (float*)alloc((size_t)64 * 4 * sizeof(float));
    float* w1p_sa2 = (float*)alloc((size_t)128 * 132 * sizeof(float));
    float* scAll   = (float*)alloc((size_t)2560 * sizeof(float));
    float* biAll   = (float*)alloc((size_t)2560 * sizeof(float));

    BNPack bp;
    for (int L = 0; L < 9; ++L)
        bp.l[L] = BNSrc{G[L], Bb[L], RM[L], RV[L], couts[L], offs[L]};

    auto LP = [&](int L, const float* wsel) {
        LayerP p;
        p.w = wsel;
        p.sc = scAll + offs[L];
        p.bi = biAll + offs[L];
        return p;
    };
    LayerP sa10 = LP(0, w1p_sa1), sa11 = LP(1, W[1]), sa12 = LP(2, W[2]);
    LayerP sa20 = LP(3, w1p_sa2), sa21 = LP(4, W[4]), sa22 = LP(5, W[5]);
    LayerP lo0  = LP(6, W[6]),    lo1  = LP(7, W[7]), lo2  = LP(8, W[8]);

    hipMemsetAsync(feat1, 0, (size_t)16 * 512 * 128 * sizeof(float), stream);
    hipMemsetAsync(feat2, 0, (size_t)16 * 128 * 256 * sizeof(float), stream);
    hipMemsetAsync(pmax, 0, (size_t)16 * 1024 * 8, stream);

    prep_kernel<<<(2560 + 256 + 128 * 132 + 255) / 256, 256, 0, stream>>>(
        bp, scAll, biAll, W[0], W[3], w1p_sa1, w1p_sa2);
    transpose_kernel<<<512, 256, 0, stream>>>(points, xyz);
    fps_kernel<<<16, 1024, 0, stream>>>(xyz, centers);
    ballquery_kernel<<<(16 * 512 + 3) / 4, 128, 0, stream>>>(
        xyz, 8192, centers, 512, 512, 0.2f * 0.2f, 32, idx1, 16);
    sa1_kernel<<<16 * 512, 128, 0, stream>>>(xyz, centers, idx1, feat1, sa10, sa11, sa12);
    ballquery_kernel<<<(16 * 128 + 3) / 4, 128, 0, stream>>>(
        centers, 512, centers, 128, 512, 0.4f * 0.4f, 64, idx2, 16);
    sa2_kernel<<<16 * 128 * 2, 128, 0, stream>>>(centers, idx2, feat1, feat2, sa20, sa21, sa22);
    local_kernel<<<16 * 8, 256, 0, stream>>>(feat2, pmax, lo0, lo1, lo2);
    finalize_kernel<<<64, 256, 0, stream>>>(pmax, (float*)d_out, (int*)d_out + 16 * 1024);
}